// PairwiseMaxMarginHingeLossFunc_21775484190872
// MI455X (gfx1250) — compile-verified
//
#include <hip/hip_runtime.h>

typedef float v2f __attribute__((ext_vector_type(2)));
typedef float v4f __attribute__((ext_vector_type(4)));
typedef float v8f __attribute__((ext_vector_type(8)));

#define N_ITEMS 2048
#define BATCH   32
#define TILES   (N_ITEMS / 16)   // 128 tile rows/cols
#define ROWGRPS 16               // tile-row groups per batch (8 waves each)
#define CHAINS  4                // independent accumulator chains (ILP)

// One block = one (batch, tile-row-group). 256 threads = 8 wave32s.
// Rank-4 pair matrix with margin folded into the K=1 row of B:
//   D = A x B' + C = (margin - P) + C
// hinge+accumulate as a max-recurrence:
//   C_t = max(D_t, C_{t-1})  ==  C_{t-1} + max(margin - P_t, 0)
// Both the WMMA and the max are emitted as raw asm: the WMMA so the register
// allocator uses the untied VDST!=SRC2 form (no acc->D copies), the max so
// LLVM cannot insert canonicalize passes.
__global__ __launch_bounds__(256) void pairwise_hinge_wmma_kernel(
    const float* __restrict__ outp,   // [N_ITEMS, BATCH] row-major
    const float* __restrict__ labp,   // [N_ITEMS, BATCH] row-major
    const float* __restrict__ marginp,
    float* __restrict__ partials)     // [BATCH * ROWGRPS]
{
    // AoS record {o, l, a=o*l, margin-a}: one ds_load_b128 per fragment. 32 KB.
    __shared__ v4f  s_v[N_ITEMS];
    __shared__ float s_red[8];

    const int tid = threadIdx.x;
    const int b   = blockIdx.y;
    const float margin = marginp[0];

    // Stage batch-b column vectors into LDS (stride-BATCH gather).
#pragma unroll
    for (int r = 0; r < N_ITEMS / 256; ++r) {
        int i = tid + r * 256;
        float o = outp[i * BATCH + b];
        float l = labp[i * BATCH + b];
        float a = o * l;
        v4f rec;
        rec.x = o; rec.y = l; rec.z = a; rec.w = margin - a;
        s_v[i] = rec;
    }
    __syncthreads();

    const int lane = tid & 31;
    const int wave = tid >> 5;
    const bool hi  = lane >= 16;     // upper half-wave holds K=2,3 rows/cols
    const int sub  = lane & 15;

    // A fragment (16x4, f32): loop-invariant for this wave's tile-row strip.
    // lanes 0-15:  A[m][0]=a_i, A[m][1]=1   (m = lane)
    // lanes 16-31: A[m][2]=o_i, A[m][3]=l_i (m = lane-16)
    const int i = (blockIdx.x * 8 + wave) * 16 + sub;
    const v4f ri = s_v[i];           // {o_i, l_i, a_i, margin-a_i}
    v2f afrag;
    afrag.x = hi ? ri.x : ri.z;
    afrag.y = hi ? ri.y : 1.0f;

    // CHAINS independent hinge-sum accumulators (tile columns interleaved).
    v8f acc[CHAINS];
#pragma unroll
    for (int u = 0; u < CHAINS; ++u)
#pragma unroll
        for (int k = 0; k < 8; ++k) acc[u][k] = 0.0f;

    for (int tj = 0; tj < TILES; tj += CHAINS) {
#pragma unroll
        for (int u = 0; u < CHAINS; ++u) {
            // B' fragment (4x16, f32): B' = [-1; (margin-a)^T; l^T; o^T]
            // lanes 0-15:  B'[0][n]=-1,  B'[1][n]=margin-a_j  (n = lane)
            // lanes 16-31: B'[2][n]=l_j, B'[3][n]=o_j         (n = lane-16)
            const int j = (tj + u) * 16 + sub;
            const v4f rj = s_v[j];   // {o_j, l_j, a_j, margin-a_j}
            v2f bfrag;
            bfrag.x = hi ? rj.y : -1.0f;
            bfrag.y = hi ? rj.x : rj.w;

            // D = A x B' + acc = (margin - P_tile) + acc   (untied VDST)
            v8f d;
            asm("v_wmma_f32_16x16x4_f32 %0, %1, %2, %3"
                : "=v"(d)
                : "v"(afrag), "v"(bfrag), "v"(acc[u]));

            // acc = max(D, acc)  ==  acc + max(margin - P, 0)
#pragma unroll
            for (int k = 0; k < 8; ++k) {
                float m;
                asm("v_max_num_f32 %0, %1, %2"
                    : "=v"(m) : "v"(d[k]), "v"(acc[u][k]));
                acc[u][k] = m;
            }
        }
    }

    // Combine the chains, then lanes, then waves (fixed order: deterministic).
    float sum = 0.0f;
#pragma unroll
    for (int u = 0; u < CHAINS; ++u) {
        float t = ((acc[u][0] + acc[u][1]) + (acc[u][2] + acc[u][3])) +
                  ((acc[u][4] + acc[u][5]) + (acc[u][6] + acc[u][7]));
        sum += t;
    }

#pragma unroll
    for (int off = 16; off > 0; off >>= 1)
        sum += __shfl_down(sum, off, 32);

    if (lane == 0) s_red[wave] = sum;
    __syncthreads();
    if (tid == 0) {
        float t = 0.f;
#pragma unroll
        for (int w = 0; w < 8; ++w) t += s_red[w];
        partials[b * ROWGRPS + blockIdx.x] = t;
    }
}

// Single-block deterministic final reduction over BATCH*ROWGRPS = 512 partials.
__global__ __launch_bounds__(256) void pairwise_hinge_reduce_kernel(
    const float* __restrict__ partials, float* __restrict__ out)
{
    __shared__ float s[256];
    const int tid = threadIdx.x;
    s[tid] = partials[tid] + partials[tid + 256];
    __syncthreads();
#pragma unroll
    for (int off = 128; off > 0; off >>= 1) {
        if (tid < off) s[tid] += s[tid + off];
        __syncthreads();
    }
    if (tid == 0) out[0] = s[0] * (1.0f / (2.0f * (float)N_ITEMS));
}

extern "C" void kernel_launch(void* const* d_in, const int* in_sizes, int n_in,
                              void* d_out, int out_size, void* d_ws, size_t ws_size,
                              hipStream_t stream) {
    const float* outp = (const float*)d_in[0];   // [2048, 32] f32
    const float* labp = (const float*)d_in[1];   // [2048, 32] f32
    const float* marg = (const float*)d_in[2];   // scalar f32
    float* partials = (float*)d_ws;              // 512 floats

    dim3 grid(ROWGRPS, BATCH);                   // 16 x 32 = 512 blocks
    pairwise_hinge_wmma_kernel<<<grid, 256, 0, stream>>>(outp, labp, marg, partials);
    pairwise_hinge_reduce_kernel<<<1, 256, 0, stream>>>(partials, (float*)d_out);
}